// GAT_85676007621256
// MI455X (gfx1250) — compile-verified
//
#include <hip/hip_runtime.h>
#include <hip/hip_bf16.h>
#include <cstddef>

// ---- problem constants (from reference) ----
#define N0_  100000
#define N1_  50000
#define N2_  25000
#define DEG_ 16
#define IN_  256
#define HID_ 128
#define OUT_ 128
#define H1_  2
#define H2_  1

typedef float v2f __attribute__((ext_vector_type(2)));
typedef float v8f __attribute__((ext_vector_type(8)));

// ============================================================================
// GEMM: D[M,N] = A[M,K] @ B[K,N], f32, via V_WMMA_F32_16X16X4_F32.
// N,K are template constants -> all B/A offsets become immediate offsets
// (kb*N*4 <= 256KB fits the 24-bit signed IOFFSET), no per-iter address math.
// One wave -> 16 rows x 64 cols (4 accumulator tiles), K stepped by 4.
// A fragment: lanes 0-15 hold row M=l15 {k,k+1}; lanes 16-31 hold {k+2,k+3}.
// B fragment: lanes 0-15 hold col N=l15 {k,k+1}; lanes 16-31 hold {k+2,k+3}.
// C/D: VGPR r -> M = r + (lane>=16 ? 8 : 0), N = lane&15.
// Partial last row-tile: A row index is CLAMPED (a row of A only feeds the
// same row of D, and out-of-range rows are never stored) -> unconditional
// loads, no exec-mask branching in the K loop.
// ============================================================================
template <int TN, int TK>
__global__ void gat_gemm_wmma(const float* __restrict__ A,
                              const float* __restrict__ B,
                              float* __restrict__ D,
                              int M, int tilesM) {
  constexpr int groupsN = TN >> 6;                  // 64-col groups
  const int lane = threadIdx.x & 31;
  const int wave = threadIdx.x >> 5;
  const int gid = blockIdx.x * 8 + wave;
  if (gid >= tilesM * groupsN) return;
  const int tileM = gid / groupsN;
  const int gN    = gid % groupsN;

  const int half = lane >> 4;
  const int l15  = lane & 15;
  int row = tileM * 16 + l15;
  if (row >= M) row = M - 1;                        // clamp: safe, see header
  // A row base is TK*4-byte aligned -> b64 loads are aligned
  const float2* ap = (const float2*)(A + (size_t)row * TK) + half;
  const float*  bp = B + (size_t)(half * 2) * TN + gN * 64 + l15;

  v8f c0 = {}, c1 = {}, c2 = {}, c3 = {};

#pragma unroll 8
  for (int k0 = 0; k0 < TK; k0 += 4) {
    const float2 a2 = *ap;                          // global_load_b64
    ap += 2;
    v2f a = { a2.x, a2.y };
    v2f b0 = { bp[ 0], bp[TN +  0] };
    v2f b1 = { bp[16], bp[TN + 16] };
    v2f b2 = { bp[32], bp[TN + 32] };
    v2f b3 = { bp[48], bp[TN + 48] };
    bp += 4 * TN;
    c0 = __builtin_amdgcn_wmma_f32_16x16x4_f32(false, a, false, b0, (short)0, c0, false, false);
    c1 = __builtin_amdgcn_wmma_f32_16x16x4_f32(false, a, false, b1, (short)0, c1, false, false);
    c2 = __builtin_amdgcn_wmma_f32_16x16x4_f32(false, a, false, b2, (short)0, c2, false, false);
    c3 = __builtin_amdgcn_wmma_f32_16x16x4_f32(false, a, false, b3, (short)0, c3, false, false);
  }

#pragma unroll
  for (int r = 0; r < 8; ++r) {
    const int m = tileM * 16 + r + half * 8;
    if (m < M) {
      float* Dr = D + (size_t)m * TN + gN * 64 + l15;
      Dr[0]  = c0[r];
      Dr[16] = c1[r];
      Dr[32] = c2[r];
      Dr[48] = c3[r];
    }
  }
}

// ============================================================================
// a[n,h] = dot(F[n, h*128 : (h+1)*128], att[h, :128]).  One wave per (n,h).
// ============================================================================
__global__ void gat_att_dot(const float* __restrict__ F,
                            const float* __restrict__ att,
                            float* __restrict__ out, int Nn, int H) {
  const int wid  = blockIdx.x * 8 + (threadIdx.x >> 5);
  const int lane = threadIdx.x & 31;
  if (wid >= Nn * H) return;
  const int n = wid / H, h = wid % H;
  const float4 fv = *((const float4*)(F + ((size_t)n * H + h) * 128) + lane);
  const float4 av = *((const float4*)(att + (size_t)h * 128) + lane);
  float s = fv.x * av.x + fv.y * av.y + fv.z * av.z + fv.w * av.w;
#pragma unroll
  for (int off = 16; off > 0; off >>= 1) s += __shfl_xor(s, off, 32);
  if (lane == 0) out[wid] = s;
}

// ============================================================================
// Edge-softmax aggregation. One wave per (dst, head); lanes split 128 channels.
// Edges for dst i are contiguous: i*DEG .. i*DEG+15 (row = repeat(arange,DEG)).
// mode 1: += b, BatchNorm (running stats), leaky_relu(0.01) -> OUT[Nd, H*128]
// mode 2: mean over heads (H==1) + b                        -> OUT[Nd, 128]
// ============================================================================
__global__ void gat_edge_softmax_agg(const float* __restrict__ HS,
                                     const float* __restrict__ AS,
                                     const float* __restrict__ AD,
                                     const int* __restrict__ col,
                                     const int* __restrict__ tt,
                                     const int* __restrict__ timeP,
                                     const int* __restrict__ intervalP,
                                     const float* __restrict__ b,
                                     const float* __restrict__ g,
                                     const float* __restrict__ beta,
                                     const float* __restrict__ rm,
                                     const float* __restrict__ rv,
                                     float* __restrict__ OUT,
                                     int Nd, int H, int mode) {
  const int wid  = blockIdx.x * 8 + (threadIdx.x >> 5);
  const int lane = threadIdx.x & 31;
  if (wid >= Nd * H) return;
  const int i = wid / H, h = wid % H;

  const int t0  = timeP[0];
  const int t1e = t0 + intervalP[0];
  const float adv = AD[(size_t)i * H + h];

  float e[DEG_];
  int   cj[DEG_];
  bool  val[DEG_];
  float m = -__builtin_inff();
#pragma unroll
  for (int j = 0; j < DEG_; ++j) {
    const int eidx = i * DEG_ + j;
    cj[j] = col[eidx];
    const int tv = tt[eidx];
    val[j] = (tv >= t0) && (tv < t1e);
    float ev = AS[(size_t)cj[j] * H + h] + adv;
    ev = ev > 0.0f ? ev : 0.2f * ev;          // leaky_relu slope 0.2
    e[j] = ev;
    if (val[j] && ev > m) m = ev;
  }
  if (!(m > -__builtin_inff())) m = 0.0f;     // no valid edges -> m := 0

  float w[DEG_];
  float denom = 0.0f;
#pragma unroll
  for (int j = 0; j < DEG_; ++j) {
    const float ex = val[j] ? __expf(e[j] - m) : 0.0f;
    w[j] = ex;
    denom += ex;
  }
  const float inv = 1.0f / fmaxf(denom, 1e-16f);

  float4 acc = {0.0f, 0.0f, 0.0f, 0.0f};
#pragma unroll
  for (int j = 0; j < DEG_; ++j) {
    if (w[j] != 0.0f) {
      const float4 hv = *((const float4*)(HS + ((size_t)cj[j] * H + h) * 128) + lane);
      const float wj = w[j] * inv;
      acc.x += wj * hv.x; acc.y += wj * hv.y;
      acc.z += wj * hv.z; acc.w += wj * hv.w;
    }
  }

  float vals[4] = {acc.x, acc.y, acc.z, acc.w};
  if (mode == 1) {
#pragma unroll
    for (int q = 0; q < 4; ++q) {
      const int cc = h * 128 + lane * 4 + q;
      float vv = vals[q] + b[cc];
      vv = (vv - rm[cc]) * rsqrtf(rv[cc] + 1e-5f) * g[cc] + beta[cc];
      vv = vv > 0.0f ? vv : 0.01f * vv;       // leaky_relu slope 0.01
      OUT[(size_t)i * (H * 128) + cc] = vv;
    }
  } else {
#pragma unroll
    for (int q = 0; q < 4; ++q) {
      const int cc = lane * 4 + q;
      OUT[(size_t)i * 128 + cc] = vals[q] / (float)H + b[cc];
    }
  }
}

static inline int cdiv(int a, int b) { return (a + b - 1) / b; }

extern "C" void kernel_launch(void* const* d_in, const int* in_sizes, int n_in,
                              void* d_out, int out_size, void* d_ws, size_t ws_size,
                              hipStream_t stream) {
  const float* x     = (const float*)d_in[0];
  const int*   col1  = (const int*)d_in[2];
  const int*   t1    = (const int*)d_in[3];
  const int*   col2  = (const int*)d_in[5];
  const int*   t2    = (const int*)d_in[6];
  const int*   timeP = (const int*)d_in[7];
  const int*   intvP = (const int*)d_in[8];
  const float* Ws1   = (const float*)d_in[9];
  const float* Wd1   = (const float*)d_in[10];
  const float* as1   = (const float*)d_in[11];
  const float* ad1   = (const float*)d_in[12];
  const float* b1    = (const float*)d_in[13];
  const float* g1    = (const float*)d_in[14];
  const float* beta1 = (const float*)d_in[15];
  const float* m1    = (const float*)d_in[16];
  const float* v1    = (const float*)d_in[17];
  const float* Ws2   = (const float*)d_in[18];
  const float* Wd2   = (const float*)d_in[19];
  const float* as2   = (const float*)d_in[20];
  const float* ad2   = (const float*)d_in[21];
  const float* b2    = (const float*)d_in[22];

  // ---- workspace layout (floats) ----
  float* ws   = (float*)d_ws;
  float* hs1  = ws;                                   // N0*256
  float* hd1  = hs1  + (size_t)N0_ * 256;             // N1*256 (reused below)
  float* as1v = hd1  + (size_t)N1_ * 256;             // N0*H1
  float* ad1v = as1v + (size_t)N0_ * H1_;             // N1*H1
  float* h    = ad1v + (size_t)N1_ * H1_;             // N1*256
  float* as2v = h    + (size_t)N1_ * 256;             // N1
  float* ad2v = as2v + (size_t)N1_;                   // N2
  float* hs2  = hd1;                                  // reuse hd1 region: N1*128
  float* hd2  = hd1  + (size_t)N1_ * 128;             // N2*128 (fits in hd1 region)

  const dim3 blk(256);

  // ---- layer 1 GEMMs ----
  {
    int tilesM = cdiv(N0_, 16), waves = tilesM * (256 / 64);
    gat_gemm_wmma<256, 256><<<cdiv(waves, 8), blk, 0, stream>>>(x, Ws1, hs1, N0_, tilesM);
  }
  {
    int tilesM = cdiv(N1_, 16), waves = tilesM * (256 / 64);
    gat_gemm_wmma<256, 256><<<cdiv(waves, 8), blk, 0, stream>>>(x, Wd1, hd1, N1_, tilesM);
  }
  // ---- attention scalars ----
  gat_att_dot<<<cdiv(N0_ * H1_, 8), blk, 0, stream>>>(hs1, as1, as1v, N0_, H1_);
  gat_att_dot<<<cdiv(N1_ * H1_, 8), blk, 0, stream>>>(hd1, ad1, ad1v, N1_, H1_);
  // ---- layer 1 edge softmax + aggregate + bias/BN/lrelu ----
  gat_edge_softmax_agg<<<cdiv(N1_ * H1_, 8), blk, 0, stream>>>(
      hs1, as1v, ad1v, col1, t1, timeP, intvP,
      b1, g1, beta1, m1, v1, h, N1_, H1_, 1);

  // ---- layer 2 GEMMs ----
  {
    int tilesM = cdiv(N1_, 16), waves = tilesM * (128 / 64);
    gat_gemm_wmma<128, 256><<<cdiv(waves, 8), blk, 0, stream>>>(h, Ws2, hs2, N1_, tilesM);
  }
  {
    int tilesM = cdiv(N2_, 16), waves = tilesM * (128 / 64);
    gat_gemm_wmma<128, 256><<<cdiv(waves, 8), blk, 0, stream>>>(h, Wd2, hd2, N2_, tilesM);
  }
  // ---- attention scalars ----
  gat_att_dot<<<cdiv(N1_ * H2_, 8), blk, 0, stream>>>(hs2, as2, as2v, N1_, H2_);
  gat_att_dot<<<cdiv(N2_ * H2_, 8), blk, 0, stream>>>(hd2, ad2, ad2v, N2_, H2_);
  // ---- layer 2 edge softmax + aggregate + mean-head + bias -> d_out ----
  gat_edge_softmax_agg<<<cdiv(N2_ * H2_, 8), blk, 0, stream>>>(
      hs2, as2v, ad2v, col2, t2, timeP, intvP,
      b2, nullptr, nullptr, nullptr, nullptr, (float*)d_out, N2_, H2_, 2);
}